// convDeepSet_46437186404669
// MI455X (gfx1250) — compile-verified
//
#include <hip/hip_runtime.h>
#include <hip/hip_bf16.h>

// Problem constants (match reference)
#define B_  2
#define C_  32
#define CP_ 33      // C + density channel
#define W_  256
#define H_  256
#define X_  512
#define Y_  512

typedef __attribute__((ext_vector_type(2))) float v2f;
typedef __attribute__((ext_vector_type(8))) float v8f;

// ---------------------------------------------------------------------------
// FP32 WMMA: D(16x16) = A(16x4) x B(4x16) + C   (V_WMMA_F32_16X16X4_F32)
// ---------------------------------------------------------------------------
__device__ __forceinline__ v8f wmma4(v2f a, v2f b, v8f c) {
    // 8 args: (neg_a, A, neg_b, B, c_mod, C, reuse_a, reuse_b)
    return __builtin_amdgcn_wmma_f32_16x16x4_f32(
        false, a, false, b, (short)0, c, false, false);
}

// ---------------------------------------------------------------------------
// 32x32 output tile GEMM core (one wave32).  Row-major A[M,K], B[K,N], D[M,N].
// densPlane != nullptr -> fused clip+divide epilogue (deep-set normalization).
// ---------------------------------------------------------------------------
__device__ void gemm_tile32(const float* __restrict__ A, int lda,
                            const float* __restrict__ Bm, int ldb,
                            float* __restrict__ D, int ldd,
                            const float* __restrict__ densPlane,
                            int K, int mbase, int nbase)
{
    const int lane = threadIdx.x & 31;
    const int half = lane >> 4;   // 0 or 1: selects K pair {0,1} vs {2,3}
    const int l15  = lane & 15;

    v8f acc00 = {}, acc01 = {}, acc10 = {}, acc11 = {};

    // A fragment rows for the two M sub-tiles (per ISA 16x4 f32 layout)
    const float* Arow0 = A + (long)(mbase + l15)      * lda + half * 2;
    const float* Arow1 = A + (long)(mbase + 16 + l15) * lda + half * 2;
    // B fragment columns for the two N sub-tiles
    const int bcol0 = nbase + l15;
    const int bcol1 = nbase + 16 + l15;

#pragma unroll 4
    for (int k = 0; k < K; k += 4) {
        // A: contiguous K-pair -> 8-byte aligned float2 loads
        float2 a0p = *reinterpret_cast<const float2*>(Arow0 + k);
        float2 a1p = *reinterpret_cast<const float2*>(Arow1 + k);
        v2f a0; a0.x = a0p.x; a0.y = a0p.y;
        v2f a1; a1.x = a1p.x; a1.y = a1p.y;

        // B: K rows strided by ldb (per ISA 4x16 f32 layout)
        const int krow = k + half * 2;
        const float* Bk0 = Bm + (long)krow       * ldb;
        const float* Bk1 = Bm + (long)(krow + 1) * ldb;
        v2f b0; b0.x = Bk0[bcol0]; b0.y = Bk1[bcol0];
        v2f b1; b1.x = Bk0[bcol1]; b1.y = Bk1[bcol1];

        acc00 = wmma4(a0, b0, acc00);
        acc01 = wmma4(a0, b1, acc01);
        acc10 = wmma4(a1, b0, acc10);
        acc11 = wmma4(a1, b1, acc11);
    }

    // Store per ISA C/D layout: VGPR i, lane l -> row = i + 8*(l>>4), col = l&15
#pragma unroll
    for (int mt = 0; mt < 2; ++mt) {
        const v8f* accRow = (mt == 0) ? &acc00 : &acc10;
#pragma unroll
        for (int nt = 0; nt < 2; ++nt) {
            v8f acc = (nt == 0) ? accRow[0] : ((mt == 0) ? acc01 : acc11);
            const int col = nbase + nt * 16 + l15;
#pragma unroll
            for (int i = 0; i < 8; ++i) {
                const int row = mbase + mt * 16 + i + half * 8;
                float v = acc[i];
                if (densPlane) {
                    float dv = densPlane[(long)row * ldd + col];
                    dv = fminf(fmaxf(dv, 1e-6f), 1e5f);
                    v = v / dv;
                }
                D[(long)row * ldd + col] = v;
            }
        }
    }
}

// ---------------------------------------------------------------------------
// Prep: wt[B,C,W,H] -> wt_full[B,CP,W,H] (c==0: density, else nan_to_num)
// ---------------------------------------------------------------------------
__global__ void prep_wt_kernel(const float* __restrict__ wt,
                               float* __restrict__ wtf)
{
    const long planeWH = (long)W_ * H_;
    const long total   = (long)B_ * CP_ * planeWH;
    long idx = (long)blockIdx.x * blockDim.x + threadIdx.x;
    if (idx >= total) return;
    long p  = idx % planeWH;
    long bc = idx / planeWH;
    int  c  = (int)(bc % CP_);
    int  b  = (int)(bc / CP_);
    float v;
    if (c == 0) {
        float x = wt[((long)b * C_) * planeWH + p];
        v = (x != x) ? 0.0f : 1.0f;                 // density from NaN mask
    } else {
        float x = wt[((long)b * C_ + (c - 1)) * planeWH + p];
        v = (x != x) ? 0.0f : x;                    // nan_to_num
    }
    wtf[idx] = v;
}

// ---------------------------------------------------------------------------
// RBF weights: exp(-0.5*(xin-xout)^2/ls^2)
// outMajor=1 -> dst[b, o, i]  (w_lon transposed, ready as GEMM A)
// outMajor=0 -> dst[b, i, o]  (w_lat, ready as GEMM B)
// ---------------------------------------------------------------------------
__global__ void rbf_kernel(const float* __restrict__ xin,
                           const float* __restrict__ xout,
                           const float* __restrict__ lsp,
                           float* __restrict__ dst,
                           int Nin, int Nout, int outMajor)
{
    long total = (long)B_ * Nin * Nout;
    long idx = (long)blockIdx.x * blockDim.x + threadIdx.x;
    if (idx >= total) return;
    float ls  = lsp[0];
    float inv = 0.5f / (ls * ls);
    int b, i, o;
    if (outMajor) {
        i = (int)(idx % Nin);  long t = idx / Nin;  o = (int)(t % Nout); b = (int)(t / Nout);
    } else {
        o = (int)(idx % Nout); long t = idx / Nout; i = (int)(t % Nin);  b = (int)(t / Nin);
    }
    float d = xin[(long)b * Nin + i] - xout[(long)b * Nout + o];
    dst[idx] = expf(-d * d * inv);
}

// ---------------------------------------------------------------------------
// GEMM1: T[b,c][X,H] = w_lon_t[b][X,W] x wt_full[b,c][W,H]
// ---------------------------------------------------------------------------
__global__ void gemm1_kernel(const float* __restrict__ wlon,
                             const float* __restrict__ wtf,
                             float* __restrict__ T)
{
    const int z = blockIdx.z;            // b*CP + c
    const int b = z / CP_;
    gemm_tile32(wlon + (long)b * X_ * W_, W_,
                wtf  + (long)z * W_ * H_, H_,
                T    + (long)z * X_ * H_, H_,
                nullptr, W_,
                blockIdx.x * 32, blockIdx.y * 32);
}

// ---------------------------------------------------------------------------
// GEMM2: out[b,c][X,Y] = T[b,c][X,H] x w_lat[b][H,Y]
// cstart==0: density channel only (raw store).
// cstart==1: channels 1..C, fused divide by clipped density (already in out).
// ---------------------------------------------------------------------------
__global__ void gemm2_kernel(const float* __restrict__ T,
                             const float* __restrict__ wlat,
                             float* __restrict__ out,
                             int cstart)
{
    const int z = blockIdx.z;
    int b, c;
    if (cstart == 0) { b = z; c = 0; }
    else             { b = z / C_; c = 1 + (z % C_); }
    const float* A    = T    + (long)(b * CP_ + c) * X_ * H_;
    const float* Bm   = wlat + (long)b * H_ * Y_;
    float*       D    = out  + (long)(b * CP_ + c) * X_ * Y_;
    const float* dens = (c == 0) ? nullptr
                                 : out + (long)b * CP_ * X_ * Y_;  // channel 0 plane
    gemm_tile32(A, H_, Bm, Y_, D, Y_, dens, H_,
                blockIdx.x * 32, blockIdx.y * 32);
}

// ---------------------------------------------------------------------------
extern "C" void kernel_launch(void* const* d_in, const int* in_sizes, int n_in,
                              void* d_out, int out_size, void* d_ws, size_t ws_size,
                              hipStream_t stream)
{
    const float* x_in_lon  = (const float*)d_in[0];  // [B,W]
    const float* x_in_lat  = (const float*)d_in[1];  // [B,H]
    const float* wt        = (const float*)d_in[2];  // [B,C,W,H]
    const float* x_out_lon = (const float*)d_in[3];  // [B,X]
    const float* x_out_lat = (const float*)d_in[4];  // [B,Y]
    const float* init_ls   = (const float*)d_in[5];  // [1]
    float* out = (float*)d_out;                      // [B,CP,X,Y]
    float* ws  = (float*)d_ws;

    // Workspace layout (floats): w_lon_t | w_lat | wt_full | T   (~54 MB total)
    float* wlon = ws;                                     // [B,X,W]
    float* wlat = wlon + (long)B_ * X_ * W_;              // [B,H,Y]
    float* wtf  = wlat + (long)B_ * H_ * Y_;              // [B,CP,W,H]
    float* Tb   = wtf  + (long)B_ * CP_ * W_ * H_;        // [B,CP,X,H]

    {   // density + nan_to_num
        long n = (long)B_ * CP_ * W_ * H_;
        prep_wt_kernel<<<dim3((unsigned)((n + 255) / 256)), dim3(256), 0, stream>>>(wt, wtf);
    }
    {   // w_lon transposed: [B,X,W]
        long n = (long)B_ * X_ * W_;
        rbf_kernel<<<dim3((unsigned)((n + 255) / 256)), dim3(256), 0, stream>>>(
            x_in_lon, x_out_lon, init_ls, wlon, W_, X_, 1);
    }
    {   // w_lat: [B,H,Y]
        long n = (long)B_ * H_ * Y_;
        rbf_kernel<<<dim3((unsigned)((n + 255) / 256)), dim3(256), 0, stream>>>(
            x_in_lat, x_out_lat, init_ls, wlat, H_, Y_, 0);
    }

    // Stage 1: T = w_lon^T x wt_full  (66 planes of 512x256, K=256)
    gemm1_kernel<<<dim3(X_ / 32, H_ / 32, B_ * CP_), dim3(32), 0, stream>>>(wlon, wtf, Tb);

    // Stage 2a: density channel (raw) -> out[:,0]
    gemm2_kernel<<<dim3(X_ / 32, Y_ / 32, B_), dim3(32), 0, stream>>>(Tb, wlat, out, 0);

    // Stage 2b: data channels with fused clip+divide by density
    gemm2_kernel<<<dim3(X_ / 32, Y_ / 32, B_ * C_), dim3(32), 0, stream>>>(Tb, wlat, out, 1);
}